// PatternAwareSTGAT_94489281309
// MI455X (gfx1250) — compile-verified
//
#include <hip/hip_runtime.h>
#include <math.h>

typedef __attribute__((ext_vector_type(2))) float v2f;
typedef __attribute__((ext_vector_type(8))) float v8f;

// ---------------- helpers ----------------
__device__ __forceinline__ unsigned f2ord(float f) {
  unsigned b = __float_as_uint(f);
  return (b & 0x80000000u) ? ~b : (b | 0x80000000u);
}
__device__ __forceinline__ float ord2f(unsigned u) {
  unsigned b = (u & 0x80000000u) ? (u ^ 0x80000000u) : ~u;
  return __uint_as_float(b);
}
__device__ __forceinline__ float sigf(float x) { return 1.f / (1.f + __expf(-x)); }
#define ENC_NEG_INF 0x007FFFFFu  // f2ord(-inf)

__device__ __forceinline__ void load8f(const float* __restrict__ p, float* v) {
  float4 t0 = *(const float4*)p;
  float4 t1 = *(const float4*)(p + 4);
  v[0] = t0.x; v[1] = t0.y; v[2] = t0.z; v[3] = t0.w;
  v[4] = t1.x; v[5] = t1.y; v[6] = t1.z; v[7] = t1.w;
}
__device__ __forceinline__ void load8u(const unsigned* __restrict__ p, unsigned* v) {
  uint4 t0 = *(const uint4*)p;
  uint4 t1 = *(const uint4*)(p + 4);
  v[0] = t0.x; v[1] = t0.y; v[2] = t0.z; v[3] = t0.w;
  v[4] = t1.x; v[5] = t1.y; v[6] = t1.z; v[7] = t1.w;
}

// ---------------- GAT: feature transform via f32 WMMA ----------------
// htrans[N,32] = hin[N,KIN] @ W[KIN,32]; one 16-row tile per wave, two 16x16
// n-tiles, K stepped by 4 via V_WMMA_F32_16X16X4_F32. KIN is compile-time so
// the k-loop fully unrolls with unconditional (or cndmask-selected) loads.
template <int KIN>
__global__ void gat_gemm_wmma(const float* __restrict__ hin,
                              const float* __restrict__ W,
                              float* __restrict__ htrans, int n_nodes) {
  int wave = blockIdx.x * (blockDim.x >> 5) + (threadIdx.x >> 5);
  int lane = threadIdx.x & 31;
  int row0 = wave * 16;
  if (row0 >= n_nodes) return;

  int mrow  = min(row0 + (lane & 15), n_nodes - 1);  // clamp; store is guarded
  int khalf = (lane >> 4) * 2;      // 0 for lanes 0-15, 2 for lanes 16-31
  int col   = lane & 15;

  v8f acc0 = {};
  v8f acc1 = {};
#pragma unroll
  for (int k0 = 0; k0 < KIN; k0 += 4) {
    int ka = k0 + khalf;
    v2f a, b0, b1;
    if ((KIN & 3) == 0) {
      // contiguous pair -> single b64 load
      a[0]  = hin[mrow * KIN + ka];
      a[1]  = hin[mrow * KIN + ka + 1];
      b0[0] = W[ka * 32 + col];
      b0[1] = W[(ka + 1) * 32 + col];
      b1[0] = W[ka * 32 + 16 + col];
      b1[1] = W[(ka + 1) * 32 + 16 + col];
    } else {
      // clamp addresses, select value -> unconditional loads + v_cndmask
      int  i0 = (ka     < KIN) ? ka     : (KIN - 1);
      int  i1 = (ka + 1 < KIN) ? ka + 1 : (KIN - 1);
      bool p0 = (ka < KIN), p1 = (ka + 1 < KIN);
      float a0  = hin[mrow * KIN + i0];
      float a1  = hin[mrow * KIN + i1];
      float w00 = W[i0 * 32 + col];
      float w01 = W[i1 * 32 + col];
      float w10 = W[i0 * 32 + 16 + col];
      float w11 = W[i1 * 32 + 16 + col];
      a[0]  = p0 ? a0  : 0.f;
      a[1]  = p1 ? a1  : 0.f;
      b0[0] = p0 ? w00 : 0.f;
      b0[1] = p1 ? w01 : 0.f;
      b1[0] = p0 ? w10 : 0.f;
      b1[1] = p1 ? w11 : 0.f;
    }
    acc0 = __builtin_amdgcn_wmma_f32_16x16x4_f32(false, a, false, b0,
                                                 (short)0, acc0, false, false);
    acc1 = __builtin_amdgcn_wmma_f32_16x16x4_f32(false, a, false, b1,
                                                 (short)0, acc1, false, false);
  }
  // D layout: VGPR r -> row r (lanes 0-15) / row r+8 (lanes 16-31), col = lane&15
  int rbase = row0 + ((lane >> 4) * 8);
  if (row0 + 15 < n_nodes) {
    // full tile: single branch, unconditional clause-able stores
#pragma unroll
    for (int r = 0; r < 8; ++r) {
      int row = rbase + r;
      htrans[row * 32 + col]      = acc0[r];
      htrans[row * 32 + 16 + col] = acc1[r];
    }
  } else {
#pragma unroll
    for (int r = 0; r < 8; ++r) {
      int row = rbase + r;
      if (row < n_nodes) {
        htrans[row * 32 + col]      = acc0[r];
        htrans[row * 32 + 16 + col] = acc1[r];
      }
    }
  }
}

// ---------------- per-node attention coefficients + init ----------------
__global__ void node_prep(const float* __restrict__ htrans,
                          const float* __restrict__ a_s,
                          const float* __restrict__ a_d,
                          float* __restrict__ es, float* __restrict__ ed,
                          unsigned* __restrict__ emax, float* __restrict__ den,
                          float* __restrict__ hout, int n) {
  int i = blockIdx.x * blockDim.x + threadIdx.x;
  if (i >= n) return;
  const float* hr = htrans + (size_t)i * 32;
  float esv[8], edv[8];
#pragma unroll
  for (int q = 0; q < 8; ++q) {
    float4 t = *(const float4*)(hr + q * 4);
    float4 as = *(const float4*)(a_s + q * 4);
    float4 ad = *(const float4*)(a_d + q * 4);
    esv[q] = t.x * as.x + t.y * as.y + t.z * as.z + t.w * as.w;
    edv[q] = t.x * ad.x + t.y * ad.y + t.z * ad.z + t.w * ad.w;
  }
  float4* esp = (float4*)(es + i * 8);
  float4* edp = (float4*)(ed + i * 8);
  esp[0] = make_float4(esv[0], esv[1], esv[2], esv[3]);
  esp[1] = make_float4(esv[4], esv[5], esv[6], esv[7]);
  edp[0] = make_float4(edv[0], edv[1], edv[2], edv[3]);
  edp[1] = make_float4(edv[4], edv[5], edv[6], edv[7]);
  uint4 einit = make_uint4(ENC_NEG_INF, ENC_NEG_INF, ENC_NEG_INF, ENC_NEG_INF);
  ((uint4*)(emax + i * 8))[0] = einit;
  ((uint4*)(emax + i * 8))[1] = einit;
  float4 z = make_float4(0.f, 0.f, 0.f, 0.f);
  ((float4*)(den + i * 8))[0] = z;
  ((float4*)(den + i * 8))[1] = z;
  float4* ho = (float4*)(hout + (size_t)i * 32);
#pragma unroll
  for (int k = 0; k < 8; ++k) ho[k] = z;
}

// ---------------- edge passes ----------------
__global__ void edge_max(const int* __restrict__ src, const int* __restrict__ dst,
                         int E, int n, const float* __restrict__ es,
                         const float* __restrict__ ed, unsigned* __restrict__ emax) {
  int e = blockIdx.x * blockDim.x + threadIdx.x;
  if (e >= E + n) return;
  int s = (e < E) ? src[e] : (e - E);
  int d = (e < E) ? dst[e] : (e - E);
  float esv[8], edv[8];
  load8f(es + s * 8, esv);
  load8f(ed + d * 8, edv);
#pragma unroll
  for (int h = 0; h < 8; ++h) {
    float v = esv[h] + edv[h];
    v = (v > 0.f) ? v : 0.2f * v;
    atomicMax(&emax[d * 8 + h], f2ord(v));
  }
}

__global__ void edge_den(const int* __restrict__ src, const int* __restrict__ dst,
                         int E, int n, const float* __restrict__ es,
                         const float* __restrict__ ed,
                         const unsigned* __restrict__ emax, float* __restrict__ den) {
  int e = blockIdx.x * blockDim.x + threadIdx.x;
  if (e >= E + n) return;
  int s = (e < E) ? src[e] : (e - E);
  int d = (e < E) ? dst[e] : (e - E);
  float esv[8], edv[8];
  unsigned mv[8];
  load8f(es + s * 8, esv);
  load8f(ed + d * 8, edv);
  load8u(emax + d * 8, mv);
#pragma unroll
  for (int h = 0; h < 8; ++h) {
    float v = esv[h] + edv[h];
    v = (v > 0.f) ? v : 0.2f * v;
    atomicAdd(&den[d * 8 + h], __expf(v - ord2f(mv[h])));
  }
}

__global__ void edge_agg(const int* __restrict__ src, const int* __restrict__ dst,
                         int E, int n, const float* __restrict__ es,
                         const float* __restrict__ ed,
                         const unsigned* __restrict__ emax,
                         const float* __restrict__ den,
                         const float* __restrict__ htrans, float* __restrict__ hout) {
  int e = blockIdx.x * blockDim.x + threadIdx.x;
  if (e >= E + n) return;
  int s = (e < E) ? src[e] : (e - E);
  int d = (e < E) ? dst[e] : (e - E);
  float esv[8], edv[8], dnv[8];
  unsigned mv[8];
  load8f(es + s * 8, esv);
  load8f(ed + d * 8, edv);
  load8u(emax + d * 8, mv);
  load8f(den + d * 8, dnv);
  float hv[32];
  const float* hs = htrans + (size_t)s * 32;
#pragma unroll
  for (int q = 0; q < 8; ++q) {
    float4 t = *(const float4*)(hs + q * 4);
    hv[q * 4 + 0] = t.x; hv[q * 4 + 1] = t.y;
    hv[q * 4 + 2] = t.z; hv[q * 4 + 3] = t.w;
  }
#pragma unroll
  for (int h = 0; h < 8; ++h) {
    float v = esv[h] + edv[h];
    v = (v > 0.f) ? v : 0.2f * v;
    float a = __expf(v - ord2f(mv[h])) / (dnv[h] + 1e-16f);
#pragma unroll
    for (int c = 0; c < 4; ++c)
      atomicAdd(&hout[(size_t)d * 32 + h * 4 + c], a * hv[h * 4 + c]);
  }
}

__global__ void gat_finalize(float* __restrict__ h, const float* __restrict__ b, int n) {
  int i = blockIdx.x * blockDim.x + threadIdx.x;
  if (i >= n) return;
  float v = h[i] + b[i & 31];
  h[i] = (v > 0.f) ? v : (__expf(v) - 1.f);  // ELU
}

// ---------------- 2-layer LSTMs (trend / seasonal) ----------------
__global__ void lstm2_kernel(const float* __restrict__ xt, const float* __restrict__ xs,
    const float* tWih0, const float* tWhh0, const float* tbih0, const float* tbhh0,
    const float* tWih1, const float* tWhh1, const float* tbih1, const float* tbhh1,
    const float* sWih0, const float* sWhh0, const float* sbih0, const float* sbhh0,
    const float* sWih1, const float* sWhh1, const float* sbih1, const float* sbhh1,
    float* __restrict__ out) {
  const int TCn = 21, G = 84, SEQ = 14;
  bool iss = (blockIdx.x == 1);
  const float* x    = iss ? xs    : xt;
  const float* Wih0 = iss ? sWih0 : tWih0;
  const float* Whh0 = iss ? sWhh0 : tWhh0;
  const float* bih0 = iss ? sbih0 : tbih0;
  const float* bhh0 = iss ? sbhh0 : tbhh0;
  const float* Wih1 = iss ? sWih1 : tWih1;
  const float* Whh1 = iss ? sWhh1 : tWhh1;
  const float* bih1 = iss ? sbih1 : tbih1;
  const float* bhh1 = iss ? sbhh1 : tbhh1;

  __shared__ float h0[21], c0[21], h1[21], c1[21], g0[84], g1[84];
  int j = threadIdx.x;
  if (j < TCn) { h0[j] = 0.f; c0[j] = 0.f; h1[j] = 0.f; c1[j] = 0.f; }
  __syncthreads();

  for (int t = 0; t < SEQ; ++t) {
    float xv = x[t];
    if (j < G) {
      float g = xv * Wih0[j] + bih0[j] + bhh0[j];   // Wih0 is [84,1]
      for (int k = 0; k < TCn; ++k) g += h0[k] * Whh0[j * TCn + k];
      g0[j] = g;
    }
    __syncthreads();
    if (j < TCn) {
      float c = sigf(g0[21 + j]) * c0[j] + sigf(g0[j]) * tanhf(g0[42 + j]);
      c0[j] = c;
      h0[j] = sigf(g0[63 + j]) * tanhf(c);
    }
    __syncthreads();
    if (j < G) {
      float g = bih1[j] + bhh1[j];
      for (int k = 0; k < TCn; ++k)
        g += h0[k] * Wih1[j * TCn + k] + h1[k] * Whh1[j * TCn + k];
      g1[j] = g;
    }
    __syncthreads();
    if (j < TCn) {
      float c = sigf(g1[21 + j]) * c1[j] + sigf(g1[j]) * tanhf(g1[42 + j]);
      c1[j] = c;
      h1[j] = sigf(g1[63 + j]) * tanhf(c);
    }
    __syncthreads();
  }
  if (j < TCn) out[(iss ? 21 : 0) + j] = h1[j];
}

// ---------------- fused head ----------------
__global__ void head_kernel(const float* __restrict__ hfin, const int* __restrict__ tidx_p,
    const float* __restrict__ lstm_out, const float* __restrict__ residual,
    const float* __restrict__ cv, const float* __restrict__ resW, const float* __restrict__ resb,
    const float* __restrict__ pgW, const float* __restrict__ pgb,
    const float* __restrict__ f1W, const float* __restrict__ f1b,
    const float* __restrict__ lng, const float* __restrict__ lnb,
    const float* __restrict__ f2W, const float* __restrict__ f2b,
    const float* __restrict__ f3W, const float* __restrict__ f3b,
    float* __restrict__ out) {
  __shared__ float comb[96], gf[96], h1[64], h2[32], stats[2];
  int j = threadIdx.x;
  int tidx = *tidx_p;
  if (j < 32) comb[j] = hfin[(size_t)tidx * 32 + j];
  else if (j < 74) comb[j] = lstm_out[j - 32];
  else if (j < 96) {
    int k = j - 74;
    float v = residual[13] * resW[k] + resb[k];
    comb[j] = (v > 0.f) ? v : 0.f;
  }
  __syncthreads();
  if (j < 96) {
    float g = pgb[j];
    for (int i = 0; i < 96; ++i) g += comb[i] * pgW[i * 96 + j];
    g += cv[0] * pgW[96 * 96 + j];
    gf[j] = comb[j] * sigf(g);
  }
  __syncthreads();
  if (j < 64) {
    float v = f1b[j];
    for (int i = 0; i < 96; ++i) v += gf[i] * f1W[i * 64 + j];
    h1[j] = v;
  }
  __syncthreads();
  if (j == 0) {
    float mu = 0.f;
    for (int i = 0; i < 64; ++i) mu += h1[i];
    mu *= (1.f / 64.f);
    float var = 0.f;
    for (int i = 0; i < 64; ++i) { float d = h1[i] - mu; var += d * d; }
    var *= (1.f / 64.f);
    stats[0] = mu;
    stats[1] = rsqrtf(var + 1e-5f);
  }
  __syncthreads();
  if (j < 64) {
    float v = (h1[j] - stats[0]) * stats[1] * lng[j] + lnb[j];
    h1[j] = (v > 0.f) ? v : 0.f;
  }
  __syncthreads();
  if (j < 32) {
    float v = f2b[j];
    for (int i = 0; i < 64; ++i) v += h1[i] * f2W[i * 32 + j];
    h2[j] = (v > 0.f) ? v : 0.f;
  }
  __syncthreads();
  if (j < 14) {
    float v = f3b[j];
    for (int i = 0; i < 32; ++i) v += h2[i] * f3W[i * 14 + j];
    out[j] = v;
  }
}

// ---------------- launcher ----------------
extern "C" void kernel_launch(void* const* d_in, const int* in_sizes, int n_in,
                              void* d_out, int out_size, void* d_ws, size_t ws_size,
                              hipStream_t stream) {
  const float* x          = (const float*)d_in[0];
  const int*   edge_index = (const int*)  d_in[1];
  const float* trend      = (const float*)d_in[2];
  const float* seasonal   = (const float*)d_in[3];
  const float* residual   = (const float*)d_in[4];
  const float* cv         = (const float*)d_in[5];
  const int*   tidx       = (const int*)  d_in[6];
  const float* gat_W[3]  = {(const float*)d_in[7],  (const float*)d_in[11], (const float*)d_in[15]};
  const float* gat_as[3] = {(const float*)d_in[8],  (const float*)d_in[12], (const float*)d_in[16]};
  const float* gat_ad[3] = {(const float*)d_in[9],  (const float*)d_in[13], (const float*)d_in[17]};
  const float* gat_b[3]  = {(const float*)d_in[10], (const float*)d_in[14], (const float*)d_in[18]};
  const float* tP[8]; const float* sP[8];
  for (int k = 0; k < 8; ++k) { tP[k] = (const float*)d_in[19 + k]; sP[k] = (const float*)d_in[27 + k]; }
  const float* res_W = (const float*)d_in[35];
  const float* res_b = (const float*)d_in[36];
  const float* pg_W  = (const float*)d_in[37];
  const float* pg_b  = (const float*)d_in[38];
  const float* f1_W  = (const float*)d_in[39];
  const float* f1_b  = (const float*)d_in[40];
  const float* ln_g  = (const float*)d_in[41];
  const float* ln_b  = (const float*)d_in[42];
  const float* f2_W  = (const float*)d_in[43];
  const float* f2_b  = (const float*)d_in[44];
  const float* f3_W  = (const float*)d_in[45];
  const float* f3_b  = (const float*)d_in[46];

  int N = in_sizes[0] / 5;
  int E = in_sizes[1] / 2;
  const int* src = edge_index;
  const int* dst = edge_index + E;

  // workspace layout (floats)
  float* ws   = (float*)d_ws;
  float* P    = ws;                 // transformed features  [N,32]
  float* Q    = P + (size_t)N * 32; // hin / aggregation out [N,32]
  float* es   = Q + (size_t)N * 32; // [N,8]
  float* ed   = es + (size_t)N * 8; // [N,8]
  unsigned* emax = (unsigned*)(ed + (size_t)N * 8); // [N,8] ordered-encoded
  float* den  = (float*)(emax + (size_t)N * 8);     // [N,8]
  float* lstm_out = den + (size_t)N * 8;            // [42]

  int Etot   = E + N;
  int eblk   = (Etot + 255) / 256;
  int nblk   = (N + 127) / 128;
  int tiles  = (N + 15) / 16;
  int gblk   = (tiles + 3) / 4;     // 4 waves per 128-thread block

  const float* hin = x;
  for (int l = 0; l < 3; ++l) {
    if (l == 0)
      gat_gemm_wmma<5><<<gblk, 128, 0, stream>>>(hin, gat_W[l], P, N);
    else
      gat_gemm_wmma<32><<<gblk, 128, 0, stream>>>(hin, gat_W[l], P, N);
    node_prep<<<nblk, 128, 0, stream>>>(P, gat_as[l], gat_ad[l], es, ed, emax, den, Q, N);
    edge_max<<<eblk, 256, 0, stream>>>(src, dst, E, N, es, ed, emax);
    edge_den<<<eblk, 256, 0, stream>>>(src, dst, E, N, es, ed, emax, den);
    edge_agg<<<eblk, 256, 0, stream>>>(src, dst, E, N, es, ed, emax, den, P, Q);
    gat_finalize<<<((N * 32) + 255) / 256, 256, 0, stream>>>(Q, gat_b[l], N * 32);
    hin = Q;
  }

  lstm2_kernel<<<2, 96, 0, stream>>>(trend, seasonal,
      tP[0], tP[1], tP[2], tP[3], tP[4], tP[5], tP[6], tP[7],
      sP[0], sP[1], sP[2], sP[3], sP[4], sP[5], sP[6], sP[7], lstm_out);

  head_kernel<<<1, 128, 0, stream>>>(Q, tidx, lstm_out, residual, cv,
      res_W, res_b, pg_W, pg_b, f1_W, f1_b, ln_g, ln_b, f2_W, f2_b, f3_W, f3_b,
      (float*)d_out);
}